// End2EndRoialign_67817533603930
// MI455X (gfx1250) — compile-verified
//
#include <hip/hip_runtime.h>
#include <hip/hip_bf16.h>
#include <stdint.h>

#define N_ANCH   102000
#define FEAT     117
#define NCLS     80
#define NMC      32
#define TOPK     1000
#define MAXOBJ   100
#define PROTO_W  320
#define PROTO_PIX 102400
#define MRES     56
#define IOU_T    0.45f
#define SCORE_T  0.25f
#define MAX_WH   1280.0f

// ---- workspace layout (units of 32-bit words) ----
#define W_SCORE   0
#define W_BOXES   (W_SCORE + N_ANCH)          // 102000*4
#define W_CLS     (W_BOXES + N_ANCH*4)
#define W_HIST    (W_CLS + N_ANCH)            // 256
#define W_STATE   (W_HIST + 256)              // 8
#define W_SIDX    (W_STATE + 8)               // 1024
#define W_SSC     (W_SIDX + 1024)             // 1024
#define W_FIA     (W_SSC + 1024)              // 128
#define W_FIV     (W_FIA + 128)               // 128
#define W_BOXF    (W_FIV + 128)               // 512
#define W_COEFF   (W_BOXF + 512)              // 112*32
#define W_PROJ    (W_COEFF + 112*32)          // 112*102400 floats (~46MB, rows 100..111 pad)

typedef float v2f __attribute__((ext_vector_type(2)));
typedef float v8f __attribute__((ext_vector_type(8)));

// ---------------- decode: score, argmax class, xyxy boxes ----------------
__global__ void k_decode(const float* __restrict__ x, float* __restrict__ score,
                         float* __restrict__ boxes, unsigned* __restrict__ cls) {
  int i = blockIdx.x * blockDim.x + threadIdx.x;
  if (i >= N_ANCH) return;
  const float* p = x + (size_t)i * FEAT;
  float cx = p[0], cy = p[1], w = p[2], h = p[3], obj = p[4];
  float best = p[5]; int bi = 0;
  for (int c = 1; c < NCLS; ++c) { float v = p[5 + c]; if (v > best) { best = v; bi = c; } }
  score[i] = obj * best;
  cls[i] = (unsigned)bi;
  boxes[i * 4 + 0] = cx - w * 0.5f;
  boxes[i * 4 + 1] = cy - h * 0.5f;
  boxes[i * 4 + 2] = cx + w * 0.5f;
  boxes[i * 4 + 3] = cy + h * 0.5f;
}

__global__ void k_clear(unsigned* p, int n) {
  int i = blockIdx.x * blockDim.x + threadIdx.x;
  if (i < n) p[i] = 0u;
}

__global__ void k_init_state(unsigned* state) {
  state[0] = 0u;       // prefix
  state[1] = TOPK;     // k remaining
}

// ---- radix-select pass: histogram byte (scores >= 0 so uint order == float order)
__global__ void k_hist(const float* __restrict__ score, unsigned* __restrict__ hist,
                       const unsigned* __restrict__ state, int shift) {
  int i = blockIdx.x * blockDim.x + threadIdx.x;
  if (i >= N_ANCH) return;
  unsigned prefix = state[0];
  unsigned maskHi = (shift == 24) ? 0u : (0xFFFFFFFFu << (shift + 8));
  unsigned b = __float_as_uint(score[i]);
  if ((b & maskHi) == (prefix & maskHi))
    atomicAdd(&hist[(b >> shift) & 0xFFu], 1u);
}

__global__ void k_select(const unsigned* __restrict__ hist, unsigned* __restrict__ state, int shift) {
  unsigned k = state[1];
  unsigned cum = 0; int sel = 0;
  for (int b = 255; b >= 0; --b) {
    unsigned c = hist[b];
    if (cum + c >= k) { sel = b; break; }
    cum += c;
  }
  state[0] |= ((unsigned)sel) << shift;
  state[1] = k - cum;   // after last pass: quota among pivot-equal (lowest index first)
}

// ---------------- ordered gather of top-1000 + bitonic sort ----------------
__device__ unsigned scan_inc(unsigned v, volatile unsigned* s, int tid) {
  s[tid] = v; __syncthreads();
  for (int off = 1; off < 1024; off <<= 1) {
    unsigned t = (tid >= off) ? s[tid - off] : 0u;
    __syncthreads();
    s[tid] += t;
    __syncthreads();
  }
  return s[tid];
}

__global__ void __launch_bounds__(1024) k_gather_sort(const float* __restrict__ score,
                                                      const unsigned* __restrict__ state,
                                                      unsigned* __restrict__ sidx,
                                                      float* __restrict__ ssc) {
  __shared__ unsigned long long keys[1024];
  __shared__ unsigned s_scan[1024];
  __shared__ unsigned s_base[2];  // taken, eq
  int tid = threadIdx.x;
  keys[tid] = ~0ull;
  if (tid == 0) { s_base[0] = 0u; s_base[1] = 0u; }
  unsigned pivot = state[0];
  unsigned quota = state[1];
  __syncthreads();

  for (int base = 0; base < N_ANCH; base += 1024) {
    int i = base + tid;
    unsigned b = (i < N_ANCH) ? __float_as_uint(score[i]) : 0u;
    bool gt = (i < N_ANCH) && (b > pivot);
    bool eq = (i < N_ANCH) && (b == pivot);
    unsigned takenBase = s_base[0], eqBase = s_base[1];
    __syncthreads();

    unsigned eqf = eq ? 1u : 0u;
    unsigned eqIncl = scan_inc(eqf, s_scan, tid);
    unsigned eqTot = s_scan[1023];
    __syncthreads();

    bool take = gt || (eq && (eqBase + (eqIncl - eqf)) < quota);
    unsigned tf = take ? 1u : 0u;
    unsigned tIncl = scan_inc(tf, s_scan, tid);
    unsigned tTot = s_scan[1023];
    if (take) {
      unsigned pos = takenBase + tIncl - 1u;
      if (pos < 1024u)
        keys[pos] = (((unsigned long long)(~b)) << 32) | (unsigned long long)(unsigned)i;
    }
    __syncthreads();
    if (tid == 0) { s_base[0] = takenBase + tTot; s_base[1] = eqBase + eqTot; }
    __syncthreads();
  }

  // bitonic sort ascending: key = (~score_bits, index) -> score desc, index asc
  for (unsigned k = 2; k <= 1024; k <<= 1) {
    for (unsigned j = k >> 1; j > 0; j >>= 1) {
      __syncthreads();
      unsigned ixj = tid ^ j;
      if (ixj > (unsigned)tid) {
        bool up = ((tid & k) == 0);
        unsigned long long a = keys[tid], c = keys[ixj];
        if ((a > c) == up) { keys[tid] = c; keys[ixj] = a; }
      }
    }
  }
  __syncthreads();
  if (tid < TOPK) {
    unsigned long long kk = keys[tid];
    sidx[tid] = (unsigned)(kk & 0xFFFFFFFFull);
    ssc[tid] = __uint_as_float(~(unsigned)(kk >> 32));
  }
}

// ---------------- sequential NMS over sorted top-1000 ----------------
__global__ void __launch_bounds__(1024) k_nms(const unsigned* __restrict__ sidx,
                                              const float* __restrict__ ssc,
                                              const float* __restrict__ boxes,
                                              const unsigned* __restrict__ cls,
                                              float* __restrict__ dout,      // dets at [0..599]
                                              unsigned* __restrict__ fia,
                                              unsigned* __restrict__ fiv,
                                              float* __restrict__ boxf) {
  __shared__ float bx0[TOPK], bx1[TOPK], bx2[TOPK], bx3[TOPK], area[TOPK], sc[TOPK];
  __shared__ int keep[TOPK];
  __shared__ unsigned s_fi[MAXOBJ];
  __shared__ unsigned s_nk;
  int tid = threadIdx.x;
  if (tid < TOPK) {
    unsigned a = sidx[tid];
    float s = ssc[tid];
    float off = (float)cls[a] * MAX_WH;
    float b0 = boxes[a * 4 + 0] + off, b1 = boxes[a * 4 + 1] + off;
    float b2 = boxes[a * 4 + 2] + off, b3 = boxes[a * 4 + 3] + off;
    bx0[tid] = b0; bx1[tid] = b1; bx2[tid] = b2; bx3[tid] = b3;
    area[tid] = (b2 - b0) * (b3 - b1);
    sc[tid] = s;
    keep[tid] = (s > SCORE_T) ? 1 : 0;
  }
  __syncthreads();

  for (int i = 0; i < TOPK; ++i) {
    __syncthreads();
    if (!keep[i]) continue;       // uniform across workgroup
    if (tid > i && tid < TOPK && keep[tid]) {
      float ltx = fmaxf(bx0[i], bx0[tid]);
      float lty = fmaxf(bx1[i], bx1[tid]);
      float rbx = fminf(bx2[i], bx2[tid]);
      float rby = fminf(bx3[i], bx3[tid]);
      float w = fmaxf(rbx - ltx, 0.0f);
      float h = fmaxf(rby - lty, 0.0f);
      float inter = w * h;
      float iou = inter / (area[i] + area[tid] - inter + 1e-9f);
      if (iou > IOU_T) keep[tid] = 0;
    }
  }
  __syncthreads();

  if (tid == 0) {
    unsigned c = 0;
    for (int j = 0; j < TOPK && c < MAXOBJ; ++j)
      if (keep[j]) s_fi[c++] = (unsigned)j;
    s_nk = c;
  }
  __syncthreads();

  if (tid < MAXOBJ) {
    bool valid = (unsigned)tid < s_nk;
    unsigned j = valid ? s_fi[tid] : 0u;
    unsigned a = valid ? sidx[j] : 0u;
    float vf = valid ? 1.0f : 0.0f;
    float b0 = boxes[a * 4 + 0] * vf, b1 = boxes[a * 4 + 1] * vf;
    float b2 = boxes[a * 4 + 2] * vf, b3 = boxes[a * 4 + 3] * vf;
    dout[tid * 6 + 0] = b0;
    dout[tid * 6 + 1] = b1;
    dout[tid * 6 + 2] = b2;
    dout[tid * 6 + 3] = b3;
    dout[tid * 6 + 4] = valid ? sc[j] : 0.0f;
    dout[tid * 6 + 5] = valid ? ((float)cls[a] * vf) : 0.0f;
    fia[tid] = a;
    fiv[tid] = valid ? 1u : 0u;
    boxf[tid * 4 + 0] = b0; boxf[tid * 4 + 1] = b1;
    boxf[tid * 4 + 2] = b2; boxf[tid * 4 + 3] = b3;
  }
}

// ---------------- gather coefficients into padded 112x32 A matrix ----------------
__global__ void k_coeff(const float* __restrict__ x, const unsigned* __restrict__ fia,
                        const unsigned* __restrict__ fiv, float* __restrict__ coeff) {
  int i = blockIdx.x * blockDim.x + threadIdx.x;
  if (i >= 112 * NMC) return;
  int row = i / NMC, c = i % NMC;
  float v = 0.0f;
  if (row < MAXOBJ && fiv[row])
    v = x[(size_t)fia[row] * FEAT + 5 + NCLS + c];
  coeff[i] = v;
}

// ---------------- WMMA fp32 GEMM: proj[112,102400] = coeff[112,32] @ proto[32,102400] ----------------
// proj padded to 112 rows -> unconditional, branch-free D-tile stores.
__global__ void __launch_bounds__(256) k_gemm(const float* __restrict__ coeff,
                                              const float* __restrict__ proto,
                                              float* __restrict__ proj) {
  int wid = blockIdx.x * 8 + (threadIdx.x >> 5);   // 44800 waves = 7 Mtiles x 6400 Ntiles
  int lane = threadIdx.x & 31;
  int mt = wid % 7;
  int nt = wid / 7;
  int half = lane >> 4;       // K pair select per ISA A/B layout
  int l = lane & 15;
  int mrow = mt * 16 + l;     // A row this lane carries
  int pix = nt * 16 + l;      // B/D column this lane carries
  const float* A = coeff + mrow * NMC;
  const float* B = proto + pix;
  v8f acc = {0.f, 0.f, 0.f, 0.f, 0.f, 0.f, 0.f, 0.f};
#pragma unroll
  for (int k0 = 0; k0 < NMC; k0 += 4) {
    int ka = k0 + half * 2;
    v2f a; a.x = A[ka]; a.y = A[ka + 1];
    v2f b; b.x = B[(size_t)ka * PROTO_PIX]; b.y = B[(size_t)(ka + 1) * PROTO_PIX];
    acc = __builtin_amdgcn_wmma_f32_16x16x4_f32(false, a, false, b, (short)0, acc, false, false);
  }
  // C/D layout: vgpr r -> M = mt*16 + half*8 + r, N = pix
  float* P = proj + (size_t)(mt * 16 + half * 8) * PROTO_PIX + pix;
#pragma unroll
  for (int r = 0; r < 8; ++r)
    P[(size_t)r * PROTO_PIX] = acc[r];
}

// ---------------- fused ROI-align (on projected plane) + avgpool + sigmoid ----------------
__global__ void k_mask(const float* __restrict__ proj, const float* __restrict__ boxf,
                       const unsigned* __restrict__ fiv, float* __restrict__ outm) {
  int gid = blockIdx.x * blockDim.x + threadIdx.x;
  if (gid >= MAXOBJ * MRES * MRES) return;
  int n = gid / (MRES * MRES);
  int rem = gid % (MRES * MRES);
  int h = rem / MRES, w = rem % MRES;
  if (!fiv[n]) { outm[gid] = 0.0f; return; }
  float b0 = boxf[n * 4 + 0] * 0.25f, b1 = boxf[n * 4 + 1] * 0.25f;
  float b2 = boxf[n * 4 + 2] * 0.25f, b3 = boxf[n * 4 + 3] * 0.25f;
  float rw = fmaxf(b2 - b0, 1.0f), rh = fmaxf(b3 - b1, 1.0f);
  const float* pr = proj + (size_t)n * PROTO_PIX;
  float sum = 0.0f;
#pragma unroll
  for (int dy = 0; dy < 2; ++dy) {
    int p = 2 * h + dy;
    float ty = ((float)p + 0.5f) * (1.0f / 112.0f);
    float ys = b1 + ty * rh;
    bool vy = (ys > -1.0f) && (ys < (float)PROTO_W);
    float yc = fminf(fmaxf(ys, 0.0f), (float)(PROTO_W - 1));
    int y0 = (int)floorf(yc);
    int y1 = min(y0 + 1, PROTO_W - 1);
    float ly = yc - (float)y0;
#pragma unroll
    for (int dx = 0; dx < 2; ++dx) {
      int q = 2 * w + dx;
      float tx = ((float)q + 0.5f) * (1.0f / 112.0f);
      float xs = b0 + tx * rw;
      bool vx = (xs > -1.0f) && (xs < (float)PROTO_W);
      float xc = fminf(fmaxf(xs, 0.0f), (float)(PROTO_W - 1));
      int x0 = (int)floorf(xc);
      int x1 = min(x0 + 1, PROTO_W - 1);
      float lx = xc - (float)x0;
      float v = pr[y0 * PROTO_W + x0] * (1.0f - ly) * (1.0f - lx)
              + pr[y0 * PROTO_W + x1] * (1.0f - ly) * lx
              + pr[y1 * PROTO_W + x0] * ly * (1.0f - lx)
              + pr[y1 * PROTO_W + x1] * ly * lx;
      if (vy && vx) sum += v;
    }
  }
  float m = sum * 0.25f;
  outm[gid] = 1.0f / (1.0f + __expf(-m));
}

// ---------------- host orchestration ----------------
extern "C" void kernel_launch(void* const* d_in, const int* in_sizes, int n_in,
                              void* d_out, int out_size, void* d_ws, size_t ws_size,
                              hipStream_t stream) {
  const float* x = (const float*)d_in[0];      // (1,102000,117)
  const float* proto = (const float*)d_in[1];  // (1,32,320,320)
  float* out = (float*)d_out;                  // dets(100*6) ++ masks(100*56*56)

  float* wsf = (float*)d_ws;
  unsigned* wsu = (unsigned*)d_ws;

  float* score = wsf + W_SCORE;
  float* boxes = wsf + W_BOXES;
  unsigned* cls = wsu + W_CLS;
  unsigned* hist = wsu + W_HIST;
  unsigned* state = wsu + W_STATE;
  unsigned* sidx = wsu + W_SIDX;
  float* ssc = wsf + W_SSC;
  unsigned* fia = wsu + W_FIA;
  unsigned* fiv = wsu + W_FIV;
  float* boxf = wsf + W_BOXF;
  float* coeff = wsf + W_COEFF;
  float* proj = wsf + W_PROJ;

  // 1) decode
  k_decode<<<(N_ANCH + 255) / 256, 256, 0, stream>>>(x, score, boxes, cls);

  // 2) exact top-1000 pivot via 4-pass radix select over float bits
  k_init_state<<<1, 1, 0, stream>>>(state);
  const int shifts[4] = {24, 16, 8, 0};
  for (int p = 0; p < 4; ++p) {
    k_clear<<<1, 256, 0, stream>>>(hist, 256);
    k_hist<<<(N_ANCH + 255) / 256, 256, 0, stream>>>(score, hist, state, shifts[p]);
    k_select<<<1, 1, 0, stream>>>(hist, state, shifts[p]);
  }

  // 3) ordered gather of top-1000 + bitonic sort (score desc, index asc)
  k_gather_sort<<<1, 1024, 0, stream>>>(score, state, sidx, ssc);

  // 4) sequential NMS, select first 100 kept, emit dets
  k_nms<<<1, 1024, 0, stream>>>(sidx, ssc, boxes, cls, out, fia, fiv, boxf);

  // 5) gather coefficients (padded to 112 rows)
  k_coeff<<<(112 * NMC + 255) / 256, 256, 0, stream>>>(x, fia, fiv, coeff);

  // 6) WMMA fp32 GEMM: proj = coeff @ proto   (M=112 padded, K=32, N=102400)
  k_gemm<<<5600, 256, 0, stream>>>(coeff, proto, proj);

  // 7) fused ROI-align + 2x2 avgpool + sigmoid on projected planes
  k_mask<<<(MAXOBJ * MRES * MRES + 255) / 256, 256, 0, stream>>>(proj, boxf, fiv, out + MAXOBJ * 6);
}